// SSIMLoss_7378753815223
// MI455X (gfx1250) — compile-verified
//
#include <hip/hip_runtime.h>

typedef __attribute__((ext_vector_type(2))) float v2f;
typedef __attribute__((ext_vector_type(8))) float v8f;

#define IMG_W   1024
#define IMG_H   1024
#define TILE_W  128         // output cols per block
#define TILE_H  16          // output rows per block
#define HALO    5
#define KS      11
#define SIMG_H  32          // 16+10 rounded up to 16-row tiles
#define SIMG_W  140         // 128+10, +pad for K=28 reads (max col 139); 140%64=12
#define SH_H    32
#define SH_W    136         // 128 logical cols; 2*136%64=16 -> half-waves on disjoint banks

__global__ void __launch_bounds__(256)
ssim_wmma_kernel(const float* __restrict__ img1,
                 const float* __restrict__ img2,
                 const float* __restrict__ kern,
                 float* __restrict__ out,
                 float invN)
{
    __shared__ __align__(16) float sA[SIMG_H][SIMG_W];  // img1 tile (zero-padded)
    __shared__ __align__(16) float sB[SIMG_H][SIMG_W];  // img2 tile
    __shared__ __align__(16) float sH[SH_H][SH_W];      // row-pass intermediate
    __shared__ float sw[16];                            // 1-D gaussian
    __shared__ float sPart[8];

    const int tid  = threadIdx.x;
    const int lane = tid & 31;
    const int wv   = tid >> 5;

    // Recover 1-D gaussian g from k2d = outer(g,g):  g[i] = k2d[i][5] / sqrt(k2d[5][5])
    if (tid < KS)
        sw[tid] = kern[tid * KS + 5] * rsqrtf(kern[5 * KS + 5]);

    // Stage image tiles with zero padding (reference conv uses zero SAME padding)
    const int  gx0 = blockIdx.x * TILE_W - HALO;
    const int  gy0 = blockIdx.y * TILE_H - HALO;
    const long base = (long)blockIdx.z * (long)(IMG_W * IMG_H);
    for (int i = tid; i < SIMG_H * SIMG_W; i += 256) {
        int r = i / SIMG_W, c = i % SIMG_W;
        int gy = gy0 + r, gx = gx0 + c;
        bool ok = (gy >= 0) & (gy < IMG_H) & (gx >= 0) & (gx < IMG_W);
        long off = base + (long)gy * IMG_W + gx;
        sA[r][c] = ok ? img1[off] : 0.0f;
        sB[r][c] = ok ? img2[off] : 0.0f;
    }
    __syncthreads();

    // Banded weight fragments, used as B in the row pass and A in the col pass.
    // Chunk t covers K = 4t..4t+3.  Lane l (p = l&15, hi = 2*(l>=16)) holds
    // band(4t+hi - p) and band(4t+hi+1 - p), band(d) = (0<=d<=10) ? w[d] : 0.
    v2f wFrag[7];
    {
        int p  = lane & 15;
        int hi = (lane >> 4) << 1;
        #pragma unroll
        for (int t = 0; t < 7; ++t) {
            int k0 = 4 * t + hi - p;
            int k1 = k0 + 1;
            wFrag[t].x = (k0 >= 0 && k0 <= 10) ? sw[k0] : 0.0f;
            wFrag[t].y = (k1 >= 0 && k1 <= 10) ? sw[k1] : 0.0f;
        }
    }

    v8f acc[5];                      // conv results: mu_x, mu_y, E[xx], E[yy], E[xy]
    const int vC0 = wv * 16;         // this wave's 16x16 output tile column base

    #pragma unroll
    for (int s = 0; s < 5; ++s) {
        // ---- row pass: out_h = Sig(16x28 rows) x Band(28x16)
        // 32 rows x 128 cols of out_h = 2x8 tiles = exactly 2 tiles/wave (uniform)
        #pragma unroll
        for (int rep = 0; rep < 2; ++rep) {
            int R0 = rep * 16;                      // 0,16
            int C0 = wv * 16;                       // 0..112
            int ar = R0 + (lane & 15);              // A: lane l holds row M=l
            int cb = C0 + ((lane >> 4) << 1);       // K pair base (+2 for upper half)

            // preload only the image data this signal needs (s is compile-time)
            v2f xv[7], yv[7];
            #pragma unroll
            for (int t = 0; t < 7; ++t) {
                int ac = cb + 4 * t;                // even -> 8B-aligned v2f loads
                if (s == 0 || s == 2 || s == 4)
                    xv[t] = *(const v2f*)&sA[ar][ac];
                if (s == 1 || s == 3 || s == 4)
                    yv[t] = *(const v2f*)&sB[ar][ac];
            }

            v8f d = {};
            #pragma unroll
            for (int t = 0; t < 7; ++t) {
                v2f a;
                if      (s == 0) a = xv[t];
                else if (s == 1) a = yv[t];
                else if (s == 2) a = xv[t] * xv[t];
                else if (s == 3) a = yv[t] * yv[t];
                else             a = xv[t] * yv[t];
                d = __builtin_amdgcn_wmma_f32_16x16x4_f32(
                        false, a, false, wFrag[t], (short)0, d, false, false);
            }
            // scatter D: VGPR v -> (M = v + 8*(lane>=16), N = lane&15)
            int n  = lane & 15;
            int mo = (lane >> 4) << 3;
            #pragma unroll
            for (int v = 0; v < 8; ++v)
                sH[R0 + mo + v][C0 + n] = d[v];
        }
        __syncthreads();

        // ---- col pass: out = Band(16x28) x out_h(28x16 rows); 1 tile per wave
        {
            int n  = lane & 15;
            int jb = (lane >> 4) << 1;

            // preload all B fragments, then run the uninterrupted WMMA chain
            v2f bv[7];
            #pragma unroll
            for (int t = 0; t < 7; ++t) {
                int j = 4 * t + jb;                 // max 24+2+1 = 27 < 32
                bv[t].x = sH[j][vC0 + n];
                bv[t].y = sH[j + 1][vC0 + n];
            }

            v8f d = {};
            #pragma unroll
            for (int t = 0; t < 7; ++t)
                d = __builtin_amdgcn_wmma_f32_16x16x4_f32(
                        false, wFrag[t], false, bv[t], (short)0, d, false, false);
            acc[s] = d;
        }
        __syncthreads();                            // before next signal reuses sH
    }

    // ---- per-element SSIM + reduction
    const float C1 = 1e-4f;   // 0.01^2
    const float C2 = 9e-4f;   // 0.03^2
    float local = 0.0f;
    #pragma unroll
    for (int v = 0; v < 8; ++v) {
        float mx = acc[0][v], my = acc[1][v];
        float exx = acc[2][v], eyy = acc[3][v], exy = acc[4][v];
        float mx2 = mx * mx, my2 = my * my, mxy = mx * my;
        float sx = exx - mx2, sy = eyy - my2, sxy = exy - mxy;
        float num = (2.0f * mxy + C1) * (2.0f * sxy + C2);
        float den = (mx2 + my2 + C1) * (sx + sy + C2);
        local += num / den;
    }
    #pragma unroll
    for (int o = 16; o > 0; o >>= 1)
        local += __shfl_xor(local, o, 32);
    if (lane == 0) sPart[wv] = local;
    __syncthreads();
    if (tid == 0) {
        float t = 0.0f;
        #pragma unroll
        for (int i = 0; i < 8; ++i) t += sPart[i];
        atomicAdd(out, t * invN);
    }
}

__global__ void zero_out_kernel(float* out) { out[0] = 0.0f; }

extern "C" void kernel_launch(void* const* d_in, const int* in_sizes, int n_in,
                              void* d_out, int out_size, void* d_ws, size_t ws_size,
                              hipStream_t stream) {
    const float* img1 = (const float*)d_in[0];
    const float* img2 = (const float*)d_in[1];
    const float* kern = (const float*)d_in[2];
    float* out = (float*)d_out;

    const int batch = in_sizes[0] / (IMG_W * IMG_H);   // 16
    const float invN = 1.0f / ((float)batch * (float)IMG_W * (float)IMG_H);

    zero_out_kernel<<<1, 1, 0, stream>>>(out);

    dim3 grid(IMG_W / TILE_W, IMG_H / TILE_H, batch);  // 8 x 64 x 16
    ssim_wmma_kernel<<<grid, 256, 0, stream>>>(img1, img2, kern, out, invN);
}